// ImiRegularizeDiffusive_9887014715996
// MI455X (gfx1250) — compile-verified
//
#include <hip/hip_runtime.h>
#include <stdint.h>

// Problem dims (fixed by reference)
#define BB 2
#define CC 3
#define DD 128
#define HH 192
#define WW 192
#define SLAB (HH * WW)   // 36864 floats, one (H,W) slab
#define TP   (WW + 1)    // padded LDS row stride (193) -> conflict-free row solve

// ---- CDNA5 async global<->LDS path (probe-confirmed present) --------------
#if defined(__AMDGCN__) && __has_builtin(__builtin_amdgcn_global_load_async_to_lds_b32)
#define HAVE_ASYNC_LOAD 1
#else
#define HAVE_ASYNC_LOAD 0
#endif

#if defined(__AMDGCN__) && __has_builtin(__builtin_amdgcn_global_store_async_from_lds_b32)
#define HAVE_ASYNC_STORE 1
#else
#define HAVE_ASYNC_STORE 0
#endif

#if defined(__AMDGCN__)
typedef __attribute__((address_space(1))) int* gptr_t;  // AS1 == __device__
typedef __attribute__((address_space(3))) int* lptr_t;  // AS3 == __shared__
#endif

// Regular-temporal async load (kernel2: want L2 hits on the intermediate)
__device__ __forceinline__ void g2l_b32(const float* g, float* l) {
#if HAVE_ASYNC_LOAD
  __builtin_amdgcn_global_load_async_to_lds_b32(
      (gptr_t)(uintptr_t)g, (lptr_t)(uintptr_t)l, 0, 0);
#else
  *l = *g;
#endif
}

// Non-temporal async load (kernel1 input: read-once 113MB stream; keep it out
// of L2 so the 113MB intermediate stays resident for kernel2). gfx12 CPol:
// TH bits [2:0], TH_NT == 1.
__device__ __forceinline__ void g2l_b32_nt(const float* g, float* l) {
#if HAVE_ASYNC_LOAD
  __builtin_amdgcn_global_load_async_to_lds_b32(
      (gptr_t)(uintptr_t)g, (lptr_t)(uintptr_t)l, 0, 1 /* TH_NT */);
#else
  *l = *g;
#endif
}

__device__ __forceinline__ void l2g_b32(float* g, const float* l) {
#if HAVE_ASYNC_STORE
  __builtin_amdgcn_global_store_async_from_lds_b32(
      (gptr_t)(uintptr_t)g, (lptr_t)(uintptr_t)l, 0, 0);
#else
  *g = *l;
#endif
}

__device__ __forceinline__ void async_join() {
#if HAVE_ASYNC_LOAD || HAVE_ASYNC_STORE
#if __has_builtin(__builtin_amdgcn_s_wait_asynccnt)
  __builtin_amdgcn_s_wait_asynccnt(0);
#else
  asm volatile("s_wait_asynccnt 0" ::: "memory");
#endif
#endif
}

// ---------------------------------------------------------------------------
// Kernel 1: per (b,c,d) slab, solve along W then along H entirely in LDS.
// grid = B*C*D = 768 blocks, 192 threads each.
// Dynamic LDS: 6*192 coeff floats + 192*193 tile floats = 152,832 B
//   -> 2 workgroups per WGP (305.7 KB of 320 KB).
// ---------------------------------------------------------------------------
__global__ void __launch_bounds__(192)
solve_xy_kernel(const float* __restrict__ in,
                const float* __restrict__ ax, const float* __restrict__ bx,
                const float* __restrict__ cx,
                const float* __restrict__ ay, const float* __restrict__ by,
                const float* __restrict__ cy,
                float* __restrict__ out) {
  extern __shared__ float smem[];
  float* raX = smem;            // 192  (1/a, x axis)
  float* cXs = smem + 192;      // 191 used
  float* bXs = smem + 384;      // 191 used
  float* raY = smem + 576;      // 192
  float* cYs = smem + 768;      // 191 used
  float* bYs = smem + 960;      // 191 used
  float* tile = smem + 1152;    // HH * TP

  const int tid = threadIdx.x;           // 0..191
  const int slab = blockIdx.x;           // 0..767 (== ((b*C)+c)*D + d)
  const float* __restrict__ src = in + (size_t)slab * SLAB;
  float* __restrict__ dst = out + (size_t)slab * SLAB;

  // Coefficients + reciprocals (one division per WG per index, not per step)
  raX[tid] = 1.0f / ax[tid];
  raY[tid] = 1.0f / ay[tid];
  if (tid < WW - 1) { cXs[tid] = cx[tid]; bXs[tid] = bx[tid]; }
  if (tid < HH - 1) { cYs[tid] = cy[tid]; bYs[tid] = by[tid]; }

  // Stage the (H,W) slab: coalesced async DMA into padded LDS tile.
  // Input is read exactly once -> non-temporal so it doesn't evict the
  // intermediate (which fits in the 192MB L2) before kernel 2 reads it.
  for (int r = 0; r < HH; ++r) {
    g2l_b32_nt(&src[r * WW + tid], &tile[r * TP + tid]);
  }
  async_join();
  __syncthreads();

  // ---- X solve: thread tid owns row tid (lane LDS stride = 193 -> bank 1)
  {
    float* row = tile + tid * TP;
    float carry = row[0];
#pragma unroll 4
    for (int i = 1; i < WW; ++i) {
      carry = fmaf(-cXs[i - 1], carry, row[i]);  // y_i = x_i - c_{i-1} y_{i-1}
      row[i] = carry;
    }
    carry = row[WW - 1] * raX[WW - 1];
    row[WW - 1] = carry;
#pragma unroll 4
    for (int i = WW - 2; i >= 0; --i) {
      carry = fmaf(-bXs[i], carry, row[i]) * raX[i];  // z_i=(y_i-b_i z_{i+1})/a_i
      row[i] = carry;
    }
  }
  __syncthreads();

  // ---- Y solve: thread tid owns column tid (lane LDS stride = 1)
  {
    float carry = tile[tid];
#pragma unroll 4
    for (int j = 1; j < HH; ++j) {
      float v = fmaf(-cYs[j - 1], carry, tile[j * TP + tid]);
      tile[j * TP + tid] = v;
      carry = v;
    }
    carry = tile[(HH - 1) * TP + tid] * raY[HH - 1];
    tile[(HH - 1) * TP + tid] = carry;
#pragma unroll 4
    for (int j = HH - 2; j >= 0; --j) {
      float v = fmaf(-bYs[j], carry, tile[j * TP + tid]) * raY[j];
      tile[j * TP + tid] = v;
      carry = v;
    }
  }
  __syncthreads();

  // Store slab back (coalesced, async LDS->global, default RT so it lands in
  // L2 for kernel 2). S_ENDPGM implicitly waits on ASYNCcnt.
  for (int r = 0; r < HH; ++r) {
    l2g_b32(&dst[r * WW + tid], &tile[r * TP + tid]);
  }
}

// ---------------------------------------------------------------------------
// Kernel 2: per (b,c,h), solve along D on a (D,W) tile, in place on d_out.
// grid = B*C*H = 1152 blocks, 192 threads (one per w column).
// Dynamic LDS: 3*128 coeff + 128*192 tile = 99,840 B -> 3 WGs/WGP.
// Each thread owns its own column end-to-end -> single barrier.
// ---------------------------------------------------------------------------
__global__ void __launch_bounds__(192)
solve_z_kernel(float* __restrict__ buf,
               const float* __restrict__ az, const float* __restrict__ bz,
               const float* __restrict__ cz) {
  extern __shared__ float smem[];
  float* raZ = smem;          // 128
  float* cZs = smem + 128;    // 127 used
  float* bZs = smem + 256;    // 127 used
  float* tile = smem + 384;   // DD * WW

  const int tid = threadIdx.x;           // w in 0..191
  const int wg = blockIdx.x;             // 0..1151
  const int h = wg % HH;
  const int bc = wg / HH;                // 0..5
  const size_t base = (size_t)bc * DD * SLAB + (size_t)h * WW;

  if (tid < DD) raZ[tid] = 1.0f / az[tid];
  if (tid < DD - 1) { cZs[tid] = cz[tid]; bZs[tid] = bz[tid]; }

  // Stage (D,W) tile: perfectly coalesced per d-row; should hit L2.
  for (int d = 0; d < DD; ++d) {
    g2l_b32(&buf[base + (size_t)d * SLAB + tid], &tile[d * WW + tid]);
  }
  async_join();
  __syncthreads();  // coeff arrays were written by other waves

  // Z solve along d for column tid (lane LDS stride = 1, conflict-free)
  {
    float carry = tile[tid];
#pragma unroll 4
    for (int d = 1; d < DD; ++d) {
      float v = fmaf(-cZs[d - 1], carry, tile[d * WW + tid]);
      tile[d * WW + tid] = v;
      carry = v;
    }
    carry = tile[(DD - 1) * WW + tid] * raZ[DD - 1];
    tile[(DD - 1) * WW + tid] = carry;
#pragma unroll 4
    for (int d = DD - 2; d >= 0; --d) {
      float v = fmaf(-bZs[d], carry, tile[d * WW + tid]) * raZ[d];
      tile[d * WW + tid] = v;
      carry = v;
    }
  }

  // Store back in place (final output).
  for (int d = 0; d < DD; ++d) {
    l2g_b32(&buf[base + (size_t)d * SLAB + tid], &tile[d * WW + tid]);
  }
}

// ---------------------------------------------------------------------------
extern "C" void kernel_launch(void* const* d_in, const int* in_sizes, int n_in,
                              void* d_out, int out_size, void* d_ws, size_t ws_size,
                              hipStream_t stream) {
  const float* field = (const float*)d_in[0];
  const float* ax = (const float*)d_in[1];
  const float* bx = (const float*)d_in[2];
  const float* cx = (const float*)d_in[3];
  const float* ay = (const float*)d_in[4];
  const float* by = (const float*)d_in[5];
  const float* cy = (const float*)d_in[6];
  const float* az = (const float*)d_in[7];
  const float* bz = (const float*)d_in[8];
  const float* cz = (const float*)d_in[9];
  float* out = (float*)d_out;

  (void)in_sizes; (void)n_in; (void)out_size; (void)d_ws; (void)ws_size;

  const size_t lds1 = (size_t)(6 * 192 + HH * TP) * sizeof(float);   // 152,832 B
  const size_t lds2 = (size_t)(3 * 128 + DD * WW) * sizeof(float);   //  99,840 B

  solve_xy_kernel<<<dim3(BB * CC * DD), dim3(192), lds1, stream>>>(
      field, ax, bx, cx, ay, by, cy, out);
  solve_z_kernel<<<dim3(BB * CC * HH), dim3(192), lds2, stream>>>(
      out, az, bz, cz);
}